// DistanceLoss_14096082666214
// MI455X (gfx1250) — compile-verified
//
#include <hip/hip_runtime.h>
#include <hip/hip_bf16.h>
#include <math.h>

// MI455X / gfx1250, wave32. One workgroup per [128,128] slice.
// LDS: 2KB bitmask + 64KB g^2 array. WMMA f32 16x16x4 used for the
// per-wave reduction tree (B = ones => D[i][j] = rowsum(A_i)).

typedef float v2f __attribute__((ext_vector_type(2)));
typedef float v8f __attribute__((ext_vector_type(8)));

#define HH 128
#define WW 128
#define NSLICE 128            // B*D = 2*64
#define NPIX (HH * WW)        // 16384
#define NTHREADS 256
#define INF_F 1.0e6f
#define THRESH 0.7f
#define CLAMPV 10.0f

__global__ __launch_bounds__(NTHREADS)
void edt_slice_kernel(const float* __restrict__ y_pred,
                      const float* __restrict__ y_true,
                      float* __restrict__ S,        // [128] per-slice sum numerator
                      float* __restrict__ CNT,      // [128] per-slice nonzero(y_true) count
                      unsigned* __restrict__ FG) {  // [128] per-slice any-fg flag
    __shared__ unsigned sBits[NPIX / 32];   // 512 words: thresholded slice, 1 bit/pixel
    __shared__ float    sG2[NPIX];          // 64KB: vertical-pass squared distances
    __shared__ float    sWV[NTHREADS / 32];
    __shared__ float    sWC[NTHREADS / 32];

    const int tid  = threadIdx.x;
    const int sl   = blockIdx.x;                  // slice index in [0,128)
    const size_t base = (size_t)sl * NPIX;

    __builtin_prefetch(y_true + base, 0, 3);      // global_prefetch_b8

    // ---- Phase A: threshold y_pred, pack bits via wave32 ballot ----
    int localAny = 0;
    for (int p = tid; p < NPIX; p += NTHREADS) {
        float v = y_pred[base + p];
        int pred = (v > THRESH) ? 1 : 0;
        unsigned long long m = __ballot(pred);    // wave32: low 32 bits valid
        if ((tid & 31) == 0) sBits[p >> 5] = (unsigned)m;
        localAny |= pred;
    }
    int fgAny = __syncthreads_or(localAny);       // barrier + any-fg reduction

    // ---- Phase B: exact vertical 1D distance (both polarities), O(H) scans ----
    // At each pixel exactly one of (ga, gb) is zero, so a single float per
    // pixel (their sum / sum of squares) plus the bit encodes both.
    if (tid < WW) {
        const int j = tid;
        float ra = INF_F, rb = INF_F;             // forward running mins
        for (int i = 0; i < HH; ++i) {
            unsigned bit = (sBits[i * 4 + (j >> 5)] >> (j & 31)) & 1u;
            float pa = bit ? INF_F : 0.0f;        // pen for edt(s)
            float pb = bit ? 0.0f : INF_F;        // pen for edt(1-s)
            ra = fminf(pa, ra + 1.0f);
            rb = fminf(pb, rb + 1.0f);
            sG2[i * WW + j] = ra + rb;            // one of them is exactly 0
        }
        ra = INF_F; rb = INF_F;                   // backward running mins
        for (int i = HH - 1; i >= 0; --i) {
            unsigned bit = (sBits[i * 4 + (j >> 5)] >> (j & 31)) & 1u;
            float fwd = sG2[i * WW + j];
            float fai = bit ? fwd : 0.0f;
            float fbi = bit ? 0.0f : fwd;
            float pa = bit ? INF_F : 0.0f;
            float pb = bit ? 0.0f : INF_F;
            ra = fminf(pa, ra + 1.0f);
            rb = fminf(pb, rb + 1.0f);
            float ga = fminf(fai, ra);
            float gb = fminf(fbi, rb);
            sG2[i * WW + j] = ga * ga + gb * gb;  // exact: one term is 0
        }
    }
    __syncthreads();

    // ---- Phase C: horizontal parabola min (reference-exact O(W^2)) ----
    // ds_load_b128 fetches 4 g^2 values per DS op (broadcast address in-wave),
    // quartering DS-issue pressure vs scalar loads.
    float valSum = 0.0f, cntSum = 0.0f;
    const int j = tid & (WW - 1);
    for (int it = 0; it < NPIX / NTHREADS; ++it) {     // 64 pixels / thread
        const int i = (tid >> 7) + (it << 1);
        const float4* __restrict__ rowG4 = (const float4*)&sG2[i * WW];
        float d2a = 3.0e38f, d2b = 3.0e38f;
        float djf = (float)j;                          // j - k, k ascending
        for (int kw = 0; kw < 4; ++kw) {
            unsigned w = sBits[i * 4 + kw];
#pragma unroll
            for (int m = 0; m < 8; ++m) {
                float4 g4 = rowG4[kw * 8 + m];         // ds_load_b128
                float gv[4] = { g4.x, g4.y, g4.z, g4.w };
#pragma unroll
                for (int c = 0; c < 4; ++c) {
                    unsigned bit = (w >> (m * 4 + c)) & 1u;
                    float dj2 = djf * djf;
                    float t   = gv[c] + dj2;           // candidate when g2!=0
                    // exactly one polarity has g2 contribution here:
                    d2a = fminf(d2a, bit ? t : dj2);   // edt(s):   g2a = bit? g2 : 0
                    d2b = fminf(d2b, bit ? dj2 : t);   // edt(1-s): g2b = bit? 0 : g2
                    djf -= 1.0f;
                }
            }
        }
        float comb = fminf(sqrtf(d2a) + sqrtf(d2b), CLAMPV);
        float y = y_true[base + i * WW + j];
        valSum += comb * y;
        cntSum += (y != 0.0f) ? 1.0f : 0.0f;
    }

    // ---- Phase D: WMMA reduction (all 8 waves fully active -> EXEC all-ones) ----
    // A[i,0]=partial of lane i (rows 0..15 from lanes 0..15 as K0, lanes 16..31 as K2),
    // B = ones  =>  D[i][j] = partial(lane i) + partial(lane i+16) for all j.
    v2f aV; aV.x = valSum; aV.y = 0.0f;
    v2f aC; aC.x = cntSum; aC.y = 0.0f;
    v2f bO; bO.x = 1.0f;   bO.y = 1.0f;
    v8f cz = {};
    v8f dV = __builtin_amdgcn_wmma_f32_16x16x4_f32(false, aV, false, bO,
                                                   (short)0, cz, false, false);
    v8f dC = __builtin_amdgcn_wmma_f32_16x16x4_f32(false, aC, false, bO,
                                                   (short)0, cz, false, false);
    float sv = dV[0] + dV[1] + dV[2] + dV[3] + dV[4] + dV[5] + dV[6] + dV[7];
    float sc = dC[0] + dC[1] + dC[2] + dC[3] + dC[4] + dC[5] + dC[6] + dC[7];
    sv += __shfl_xor(sv, 16);  // combine rows 0..7 (lane n) with rows 8..15 (lane n+16)
    sc += __shfl_xor(sc, 16);
    if ((tid & 31) == 0) { sWV[tid >> 5] = sv; sWC[tid >> 5] = sc; }
    __syncthreads();
    if (tid == 0) {
        float tv = 0.0f, tc = 0.0f;
        for (int w = 0; w < NTHREADS / 32; ++w) { tv += sWV[w]; tc += sWC[w]; }
        S[sl] = tv;
        CNT[sl] = tc;
        FG[sl] = fgAny ? 1u : 0u;
    }
}

__global__ __launch_bounds__(NSLICE)
void finalize_kernel(const float* __restrict__ S,
                     const float* __restrict__ CNT,
                     const unsigned* __restrict__ FG,
                     float* __restrict__ out) {
    __shared__ int   sFirst[2], sLast[2];
    __shared__ float rv[NSLICE], rc[NSLICE];
    const int t = threadIdx.x;
    if (t < 2) { sFirst[t] = 64; sLast[t] = -1; }
    __syncthreads();
    const int b = t >> 6;       // batch
    const int d = t & 63;       // depth slice
    if (FG[t] != 0u) {
        atomicMin(&sFirst[b], d);
        atomicMax(&sLast[b], d);
    }
    __syncthreads();
    // argmax-of-all-False semantics: no fg => first=0, last=D-1 (mask everything in)
    int fb = (sFirst[b] == 64) ? 0  : sFirst[b];
    int lb = (sLast[b]  <  0) ? 63 : sLast[b];
    rv[t] = (d >= fb && d <= lb) ? S[t] : 0.0f;
    rc[t] = CNT[t];             // count_nonzero over ALL slices (unmasked)
    __syncthreads();
    for (int st = NSLICE / 2; st > 0; st >>= 1) {
        if (t < st) { rv[t] += rv[t + st]; rc[t] += rc[t + st]; }
        __syncthreads();
    }
    if (t == 0) out[0] = rv[0] / rc[0];
}

extern "C" void kernel_launch(void* const* d_in, const int* in_sizes, int n_in,
                              void* d_out, int out_size, void* d_ws, size_t ws_size,
                              hipStream_t stream) {
    (void)in_sizes; (void)n_in; (void)out_size; (void)ws_size;
    const float* y_pred = (const float*)d_in[0];
    const float* y_true = (const float*)d_in[1];
    float* out = (float*)d_out;

    float*    wsS   = (float*)d_ws;                 // [128]
    float*    wsCNT = wsS + NSLICE;                 // [128]
    unsigned* wsFG  = (unsigned*)(wsCNT + NSLICE);  // [128]

    edt_slice_kernel<<<NSLICE, NTHREADS, 0, stream>>>(y_pred, y_true, wsS, wsCNT, wsFG);
    finalize_kernel<<<1, NSLICE, 0, stream>>>(wsS, wsCNT, wsFG, out);
}